// Routing_87677462380617
// MI455X (gfx1250) — compile-verified
//
#include <hip/hip_runtime.h>
#include <math.h>

// ---------- sizes ----------
#define BATCH       64
#define IN_DIM      16
#define OUT_DIM     32
#define NIN         32768      // NUM_SHARED*H*W = 32*32*32
#define NOUT        128
#define PREDSZ      ((size_t)NIN * OUT_DIM)

typedef __attribute__((ext_vector_type(2))) float v2f;
typedef __attribute__((ext_vector_type(8))) float v8f;

// fp32 WMMA 16x16x4: D = A(16x4) * B(4x16) + C(16x16)
__device__ __forceinline__ v8f wmma_f32_4(v2f a, v2f b, v8f c) {
#if __has_builtin(__builtin_amdgcn_wmma_f32_16x16x4_f32)
    return __builtin_amdgcn_wmma_f32_16x16x4_f32(false, a, false, b, (short)0, c,
                                                 false, false);
#else
    c[0] += a.x * b.x;   // fallback so the TU still compiles
    return c;
#endif
}

// ---------- zero fill ----------
__global__ void zero_kernel(float* __restrict__ p, int n) {
    int i = blockIdx.x * blockDim.x + threadIdx.x;
    if (i < n) p[i] = 0.0f;
}

// ---------- per-sample prediction: pred[n, 0..31] = x_row(16) @ W[g](16x32) ----------
// Row n: g = n>>10, p = n&1023, channel = g*16 + (p>>6), spatial = (p&63)*16 + i
// => the 16 inputs of a row are 16 CONTIGUOUS floats in x.
__global__ void pred_kernel(const float* __restrict__ x, const float* __restrict__ W,
                            float* __restrict__ pred, int sample) {
    __shared__ float Ws[IN_DIM * OUT_DIM];          // 2 KB, one group per block
    const int n   = blockIdx.x * 256 + threadIdx.x; // 128 blocks * 256 = 32768 rows
    const int g   = (blockIdx.x * 256) >> 10;       // uniform per block (256 | 1024)
    for (int t = threadIdx.x; t < IN_DIM * OUT_DIM; t += 256)
        Ws[t] = W[g * (IN_DIM * OUT_DIM) + t];
    __syncthreads();

    const int p  = n & 1023;
    const int ch = (g << 4) + (p >> 6);
    const size_t xbase = ((size_t)sample * 512 + ch) * 1024 + (size_t)((p & 63) << 4);

    float xv[16];
    const float4* xp = reinterpret_cast<const float4*>(x + xbase);
#pragma unroll
    for (int i = 0; i < 4; ++i) {
        float4 t = xp[i];
        xv[4 * i + 0] = t.x; xv[4 * i + 1] = t.y;
        xv[4 * i + 2] = t.z; xv[4 * i + 3] = t.w;
    }

    float acc[OUT_DIM];
#pragma unroll
    for (int o = 0; o < OUT_DIM; ++o) acc[o] = 0.0f;
#pragma unroll
    for (int i = 0; i < IN_DIM; ++i) {
        const float xi = xv[i];
#pragma unroll
        for (int o = 0; o < OUT_DIM; ++o)
            acc[o] = fmaf(xi, Ws[i * OUT_DIM + o], acc[o]);
    }

    float4* op = reinterpret_cast<float4*>(pred + (size_t)n * OUT_DIM);
#pragma unroll
    for (int i = 0; i < 8; ++i)
        op[i] = make_float4(acc[4 * i + 0], acc[4 * i + 1], acc[4 * i + 2], acc[4 * i + 3]);
}

// ---------- fused scan step ----------
// Per 64-row block of b:
//   phase 1 (if do_update): b_tile += pred_prev_tile @ v^T   (WMMA, K=32, C preloaded
//            from b; result written back to global b AND into LDS as raw logits)
//            else: copy b tile global -> LDS
//   phase 2: row softmax of the LDS tile (in place)
//   phase 3: s += c_tile^T @ pred_cur_tile                   (WMMA, atomicAdd partials)
// One read + one write of b per scan step instead of two reads + one write.
__global__ void fused_step_kernel(float* __restrict__ b,
                                  const float* __restrict__ pred_prev,
                                  const float* __restrict__ v,
                                  const float* __restrict__ pred_cur,
                                  float* __restrict__ s,
                                  int do_update) {
    __shared__ float cs[64][129];   // logits -> softmax probs, padded
    __shared__ float ps[64][33];    // pred_cur tile, padded
    __shared__ float red[64][4];

    const int tid  = threadIdx.x;
    const int row0 = blockIdx.x * 64;
    const int wave = tid >> 5;
    const int lane = tid & 31;
    const int lo = lane & 15, hi = lane >> 4;

    if (do_update) {
        // ---- phase 1: WMMA b-update, 8 waves cover N=128, 4 row subtiles each ----
        const int n0 = wave * 16;
#pragma unroll
        for (int mt = 0; mt < 4; ++mt) {
            const int m0 = row0 + mt * 16;
            v8f acc;
#pragma unroll
            for (int rr = 0; rr < 8; ++rr)   // preload C = current b tile
                acc[rr] = b[(size_t)(m0 + rr + 8 * hi) * NOUT + n0 + lo];
#pragma unroll
            for (int k0 = 0; k0 < OUT_DIM; k0 += 4) {
                const int kk = k0 + hi * 2;
                v2f a;                        // A[m,k] = pred_prev[m,k]
                const float* ap = pred_prev + (size_t)(m0 + lo) * OUT_DIM + kk;
                a.x = ap[0]; a.y = ap[1];
                v2f bb;                       // B[k,n] = v[n,k]
                const float* vp = v + (size_t)(n0 + lo) * OUT_DIM + kk;
                bb.x = vp[0]; bb.y = vp[1];
                acc = wmma_f32_4(a, bb, acc);
            }
#pragma unroll
            for (int rr = 0; rr < 8; ++rr) {
                const int mg = mt * 16 + rr + 8 * hi;
                const float val = acc[rr];
                b[(size_t)(row0 + mg) * NOUT + n0 + lo] = val;  // updated logits
                cs[mg][n0 + lo] = val;                          // feed softmax
            }
        }
    } else {
        // first step: b tile straight into LDS
        for (int t = tid; t < 64 * NOUT; t += 256)
            cs[t >> 7][t & 127] = b[(size_t)(row0 + (t >> 7)) * NOUT + (t & 127)];
    }
    __syncthreads();

    // ---- phase 2: row softmax in LDS (4 threads per row) ----
    const int r = tid >> 2;      // 0..63 local row
    const int q = tid & 3;       // 32-col quarter
    float vals[32];
    float mx = -3.402823466e38f;
#pragma unroll
    for (int j = 0; j < 32; ++j) { vals[j] = cs[r][q * 32 + j]; mx = fmaxf(mx, vals[j]); }
    red[r][q] = mx;
    __syncthreads();
    const float m = fmaxf(fmaxf(red[r][0], red[r][1]), fmaxf(red[r][2], red[r][3]));
    __syncthreads();
    float sum = 0.0f;
#pragma unroll
    for (int j = 0; j < 32; ++j) { vals[j] = __expf(vals[j] - m); sum += vals[j]; }
    red[r][q] = sum;
    __syncthreads();
    const float inv = 1.0f / (red[r][0] + red[r][1] + red[r][2] + red[r][3]);
#pragma unroll
    for (int j = 0; j < 32; ++j) cs[r][q * 32 + j] = vals[j] * inv;

    // stage pred_cur tile (64 x 32)
    for (int t = tid; t < 64 * 32; t += 256)
        ps[t >> 5][t & 31] = pred_cur[(size_t)(row0 + (t >> 5)) * OUT_DIM + (t & 31)];
    __syncthreads();

    // ---- phase 3: s partial = c^T @ pred_cur (wave owns oTile, loops d tiles) ----
    const int o = wave * 16 + lo;       // A row index (M = output capsule)
    for (int dT = 0; dT < 2; ++dT) {
        const int d = dT * 16 + lo;     // B col index (N = out_dim)
        v8f acc = {};
#pragma unroll
        for (int k0 = 0; k0 < 64; k0 += 4) {
            const int kk = k0 + hi * 2; // K = local b-row n
            v2f a;  a.x = cs[kk][o];     a.y = cs[kk + 1][o];     // A[o,k] = c[k,o]
            v2f bb; bb.x = ps[kk][d];    bb.y = ps[kk + 1][d];    // B[k,d] = pred[k,d]
            acc = wmma_f32_4(a, bb, acc);
        }
#pragma unroll
        for (int rr = 0; rr < 8; ++rr) {
            const int og = wave * 16 + rr + 8 * hi;
            const int dg = dT * 16 + lo;
            atomicAdd(&s[og * OUT_DIM + dg], acc[rr]);
        }
    }
}

// ---------- squash s -> v; re-zero s; optionally emit final output ----------
__global__ void squash_kernel(float* __restrict__ s, float* __restrict__ v,
                              float* __restrict__ out, int write_out) {
    const int o = threadIdx.x;  // 128 threads
    float sv[OUT_DIM];
    float nsq = 0.0f;
#pragma unroll
    for (int k = 0; k < OUT_DIM; ++k) { sv[k] = s[o * OUT_DIM + k]; nsq = fmaf(sv[k], sv[k], nsq); }
    const float nrm   = sqrtf(nsq);
    const float coeff = nrm / (1.0f + nsq);
#pragma unroll
    for (int k = 0; k < OUT_DIM; ++k) {
        const float val = sv[k] * coeff;
        v[o * OUT_DIM + k] = val;
        if (write_out) out[o * OUT_DIM + k] = val;
        s[o * OUT_DIM + k] = 0.0f;   // ready for next step
    }
}

extern "C" void kernel_launch(void* const* d_in, const int* in_sizes, int n_in,
                              void* d_out, int out_size, void* d_ws, size_t ws_size,
                              hipStream_t stream) {
    const float* x = (const float*)d_in[0];   // [64, 512, 32, 32]
    const float* W = (const float*)d_in[1];   // [32, 16, 32]
    // d_in[2] is the zero-initialized b buffer; we keep our own copy in ws
    float* out = (float*)d_out;               // [128, 32]

    float* ws    = (float*)d_ws;
    float* b     = ws;                                  // 32768*128 = 4,194,304 f
    float* pred0 = b + (size_t)NIN * NOUT;              // ping
    float* pred1 = pred0 + PREDSZ;                      // pong
    float* s     = pred1 + PREDSZ;                      // 128*32 = 4,096 f
    float* v     = s + NOUT * OUT_DIM;                  // 128*32 = 4,096 f

    const int nB = NIN * NOUT;
    zero_kernel<<<(nB + 255) / 256, 256, 0, stream>>>(b, nB);
    zero_kernel<<<(NOUT * OUT_DIM + 255) / 256, 256, 0, stream>>>(s, NOUT * OUT_DIM);

    // step 0: no b-update, softmax(0) is uniform but computed generically
    pred_kernel<<<NIN / 256, 256, 0, stream>>>(x, W, pred0, 0);
    fused_step_kernel<<<NIN / 64, 256, 0, stream>>>(b, pred0, v, pred0, s, 0);
    squash_kernel<<<1, NOUT, 0, stream>>>(s, v, out, 0);

    for (int i = 1; i < BATCH; ++i) {
        float* cur  = (i & 1) ? pred1 : pred0;
        float* prev = (i & 1) ? pred0 : pred1;
        pred_kernel<<<NIN / 256, 256, 0, stream>>>(x, W, cur, i);
        fused_step_kernel<<<NIN / 64, 256, 0, stream>>>(b, prev, v, cur, s, 1);
        squash_kernel<<<1, NOUT, 0, stream>>>(s, v, out, (i == BATCH - 1) ? 1 : 0);
    }
}